// HeteroLinear_84619445665792
// MI455X (gfx1250) — compile-verified
//
#include <hip/hip_runtime.h>
#include <hip/hip_bf16.h>

typedef float v2f __attribute__((ext_vector_type(2)));
typedef float v4f __attribute__((ext_vector_type(4)));
typedef float v8f __attribute__((ext_vector_type(8)));

#define IN_DIM   128
#define OUT_DIM  64
#define NTYPES   8
#define CHUNK    256
#define NWAVES   8
#define MAXTILES (CHUNK / 16 + NTYPES)   // worst case sum ceil(c_t/16) = 16 + 8 = 24

// Packed B layout: Bpack[t][nt][lane][ks][v]  (v=0,1), 65536 floats = 256 KB.
//   value = W[t][4*ks + v + 2*(lane>>4)][nt*16 + (lane&15)]
// Per (t,nt,lane): 64 contiguous floats -> 16 x b128 loads in the GEMM loop.
#define PACK_FLOATS (NTYPES * 4 * 32 * 32 * 2)   // 65536
#define PACK_BYTES  (PACK_FLOATS * 4)            // 262144

__global__ __launch_bounds__(256) void pack_W_kernel(
    const float* __restrict__ W, float* __restrict__ Bpack)
{
    int idx = blockIdx.x * blockDim.x + threadIdx.x;   // 0 .. 65535
    int v    = idx & 1;
    int ks   = (idx >> 1) & 31;
    int lane = (idx >> 6) & 31;
    int nt   = (idx >> 11) & 3;
    int t    = idx >> 13;
    int k = 4 * ks + v + 2 * (lane >> 4);
    int n = nt * 16 + (lane & 15);
    Bpack[idx] = W[t * (IN_DIM * OUT_DIM) + k * OUT_DIM + n];
}

// ---------------- common device body (templated on B source) ----------------

template <bool PACKED>
__global__ __launch_bounds__(256) void hetero_linear_wmma(
    const float* __restrict__ x,          // [N, 128]
    const int*   __restrict__ node_type,  // [N]
    const float* __restrict__ W,          // [8, 128, 64]  (used when !PACKED)
    const float* __restrict__ bias,       // [8, 64]
    const float* __restrict__ Bpack,      // packed B (used when PACKED)
    float*       __restrict__ out,        // [N, 64]
    int N)
{
    __shared__ int cnt[NTYPES];
    __shared__ int lists[NTYPES][CHUNK];
    __shared__ int tType[MAXTILES];
    __shared__ int tOff[MAXTILES];
    __shared__ int totalTiles;

    const int tid  = threadIdx.x;
    const int base = blockIdx.x * CHUNK;

    // ---- Phase 1: bucket row indices by type (LDS compaction) ----
    if (tid < NTYPES) cnt[tid] = 0;
    __syncthreads();

    {
        int row = base + tid;
        if (row < N) {
            int t   = node_type[row];
            int pos = atomicAdd(&cnt[t], 1);
            lists[t][pos] = row;
        }
    }
    __syncthreads();

    // ---- Phase 2: build the tile worklist ----
    if (tid == 0) {
        int n = 0;
        for (int t = 0; t < NTYPES; ++t) {
            int c = cnt[t];
            for (int s = 0; s < c; s += 16) {
                tType[n] = t;
                tOff[n]  = s;
                ++n;
            }
        }
        totalTiles = n;
    }
    __syncthreads();

    const int wave = tid >> 5;
    const int lane = tid & 31;
    const int half = lane >> 4;   // 0: lanes 0-15, 1: lanes 16-31
    const int l16  = lane & 15;

    const int nTiles = totalTiles;

    // ---- Phase 3: each wave computes gathered 16x64 GEMM tiles ----
    // Wave-uniform loop: EXEC stays all-ones at every WMMA.
    for (int tile = wave; tile < nTiles; tile += NWAVES) {
        const int t     = tType[tile];
        const int s     = tOff[tile];
        const int count = cnt[t];

        // --- A fragments: 16 gathered rows x 128 K of x.
        // A layout (f32 16x16x4): lane holds row M = l16; k-step ks covers
        // K = 4*ks + 2*half + {0,1}.
        const int slotA = s + l16;
        const int rowA  = (slotA < count) ? lists[t][slotA] : -1;

        v2f a[32];
        if (rowA >= 0) {
            const v2f* xp = (const v2f*)(x + (long)rowA * IN_DIM + 2 * half);
            #pragma unroll
            for (int ks = 0; ks < 32; ++ks)
                a[ks] = xp[2 * ks];              // 8B-aligned b64 loads
        } else {
            #pragma unroll
            for (int ks = 0; ks < 32; ++ks) { a[ks][0] = 0.0f; a[ks][1] = 0.0f; }
        }

        // --- Accumulators seeded with bias: C(M,N) = bias[t][N] (M-invariant).
        const float* bp = bias + t * OUT_DIM;
        v8f acc[4];
        #pragma unroll
        for (int nt = 0; nt < 4; ++nt) {
            float bc = bp[nt * 16 + l16];
            #pragma unroll
            for (int v = 0; v < 8; ++v) acc[nt][v] = bc;
        }

        // --- K loop: 32 k-steps x 4 column tiles = 128 WMMAs.
        if (PACKED) {
            // Packed path: per lane, B for (t,nt) is 64 contiguous floats.
            const float* Bb = Bpack + (long)t * 8192 + lane * 64;
            const v4f* Bq0 = (const v4f*)(Bb + 0 * 2048);
            const v4f* Bq1 = (const v4f*)(Bb + 1 * 2048);
            const v4f* Bq2 = (const v4f*)(Bb + 2 * 2048);
            const v4f* Bq3 = (const v4f*)(Bb + 3 * 2048);
            #pragma unroll
            for (int kp = 0; kp < 16; ++kp) {     // two k-steps per iteration
                v4f q0 = Bq0[kp];
                v4f q1 = Bq1[kp];
                v4f q2 = Bq2[kp];
                v4f q3 = Bq3[kp];
                v2f b0l, b0h, b1l, b1h, b2l, b2h, b3l, b3h;
                b0l[0]=q0[0]; b0l[1]=q0[1]; b0h[0]=q0[2]; b0h[1]=q0[3];
                b1l[0]=q1[0]; b1l[1]=q1[1]; b1h[0]=q1[2]; b1h[1]=q1[3];
                b2l[0]=q2[0]; b2l[1]=q2[1]; b2h[0]=q2[2]; b2h[1]=q2[3];
                b3l[0]=q3[0]; b3l[1]=q3[1]; b3h[0]=q3[2]; b3h[1]=q3[3];
                // Interleave the 4 independent accumulator chains.
                acc[0] = __builtin_amdgcn_wmma_f32_16x16x4_f32(false, a[2*kp],   false, b0l, (short)0, acc[0], false, false);
                acc[1] = __builtin_amdgcn_wmma_f32_16x16x4_f32(false, a[2*kp],   false, b1l, (short)0, acc[1], false, false);
                acc[2] = __builtin_amdgcn_wmma_f32_16x16x4_f32(false, a[2*kp],   false, b2l, (short)0, acc[2], false, false);
                acc[3] = __builtin_amdgcn_wmma_f32_16x16x4_f32(false, a[2*kp],   false, b3l, (short)0, acc[3], false, false);
                acc[0] = __builtin_amdgcn_wmma_f32_16x16x4_f32(false, a[2*kp+1], false, b0h, (short)0, acc[0], false, false);
                acc[1] = __builtin_amdgcn_wmma_f32_16x16x4_f32(false, a[2*kp+1], false, b1h, (short)0, acc[1], false, false);
                acc[2] = __builtin_amdgcn_wmma_f32_16x16x4_f32(false, a[2*kp+1], false, b2h, (short)0, acc[2], false, false);
                acc[3] = __builtin_amdgcn_wmma_f32_16x16x4_f32(false, a[2*kp+1], false, b3h, (short)0, acc[3], false, false);
            }
        } else {
            // Fallback: B fragments straight from W (L2-hot, narrow loads).
            const float* Wp = W + t * (IN_DIM * OUT_DIM);
            #pragma unroll
            for (int ks = 0; ks < 32; ++ks) {
                const float* wrow = Wp + (4 * ks + 2 * half) * OUT_DIM + l16;
                #pragma unroll
                for (int nt = 0; nt < 4; ++nt) {
                    v2f bf;
                    bf[0] = wrow[nt * 16];
                    bf[1] = wrow[nt * 16 + OUT_DIM];
                    acc[nt] = __builtin_amdgcn_wmma_f32_16x16x4_f32(
                        false, a[ks], false, bf, (short)0, acc[nt], false, false);
                }
            }
        }

        // --- Scatter D: lane (L,v) holds (M = v + 8*half, N = l16).
        #pragma unroll
        for (int v = 0; v < 8; ++v) {
            const int M    = v + 8 * half;
            const int slot = s + M;
            if (slot < count) {
                const int r = lists[t][slot];
                float* op = out + (long)r * OUT_DIM + l16;
                op[0]  = acc[0][v];
                op[16] = acc[1][v];
                op[32] = acc[2][v];
                op[48] = acc[3][v];
            }
        }
    }
}

extern "C" void kernel_launch(void* const* d_in, const int* in_sizes, int n_in,
                              void* d_out, int out_size, void* d_ws, size_t ws_size,
                              hipStream_t stream) {
    const float* x         = (const float*)d_in[0];
    const int*   node_type = (const int*)  d_in[1];
    const float* W         = (const float*)d_in[2];
    const float* bias      = (const float*)d_in[3];
    float*       out       = (float*)d_out;

    const int N    = in_sizes[1];              // node_type has N elements
    const int grid = (N + CHUNK - 1) / CHUNK;

    if (ws_size >= (size_t)PACK_BYTES) {
        float* Bpack = (float*)d_ws;
        hipLaunchKernelGGL(pack_W_kernel, dim3(PACK_FLOATS / 256), dim3(256), 0, stream,
                           W, Bpack);
        hipLaunchKernelGGL((hetero_linear_wmma<true>), dim3(grid), dim3(256), 0, stream,
                           x, node_type, W, bias, Bpack, out, N);
    } else {
        hipLaunchKernelGGL((hetero_linear_wmma<false>), dim3(grid), dim3(256), 0, stream,
                           x, node_type, W, bias, (const float*)nullptr, out, N);
    }
}